// ClassificationNetwork_74818330296706
// MI455X (gfx1250) — compile-verified
//
#include <hip/hip_runtime.h>
#include <hip/hip_bf16.h>
#include <math.h>

// ---------------------------------------------------------------------------
// ROI-Align + 3-layer MLP head for MI455X (gfx1250, wave32, WMMA)
// Matrix path: V_WMMA_F32_16X16X4_F32 (full fp32, matches reference numerics;
// workload is memory-bound so low-precision WMMA buys nothing).
// ---------------------------------------------------------------------------

typedef __attribute__((ext_vector_type(2))) float v2f;
typedef __attribute__((ext_vector_type(4))) float f4;
typedef __attribute__((ext_vector_type(8))) float v8f;

#define B_    16
#define P_    22
#define H_    128
#define W_    128
#define C_    256
#define FEAT_ 4096          // 4*4*256
#define NROWS 352           // B*P  (== 22 tiles of 16 rows, exact fit)
#define NCLS  10

// ===========================================================================
// Kernel 1: ROI align -> X[NROWS][FEAT]   (one proposal per block)
//   thread t: channel-group c4q = t&63 (float4), base bin = t>>6, 4 bins each
//   fm[b,y,x,:] is contiguous (1KB) -> all 4 corner reads are coalesced.
// ===========================================================================
__global__ __launch_bounds__(256) void roi_align_kernel(
    const float* __restrict__ fm,
    const float* __restrict__ prop,
    float* __restrict__ X) {
  const int row = blockIdx.x;            // 0..351  == b*P + p
  const int b   = row / P_;
  const int t   = threadIdx.x;
  const int c4q = t & 63;                // float4 channel group 0..63
  const int bin0 = t >> 6;               // 0..3

  const float y1 = prop[row * 4 + 0];
  const float x1 = prop[row * 4 + 1];
  const float y2 = prop[row * 4 + 2];
  const float x2 = prop[row * 4 + 3];
  const float bh = (y2 - y1) * 0.25f;
  const float bw = (x2 - x1) * 0.25f;

  const f4* __restrict__ img = (const f4*)(fm) + (size_t)b * H_ * W_ * (C_ / 4);
  f4* __restrict__ Xrow = (f4*)(X + (size_t)row * FEAT_);

#pragma unroll
  for (int i = 0; i < 4; ++i) {
    const int bin = bin0 + i * 4;        // 0..15
    const float by = (float)(bin >> 2);
    const float bx = (float)(bin & 3);

    float ys = y1 + (by + 0.5f) * bh;
    float xs = x1 + (bx + 0.5f) * bw;
    ys = fminf(fmaxf(ys, 0.0f), (float)(H_ - 1));
    xs = fminf(fmaxf(xs, 0.0f), (float)(W_ - 1));

    int y0 = (int)floorf(ys);
    int x0 = (int)floorf(xs);
    y0 = min(max(y0, 0), H_ - 1);
    x0 = min(max(x0, 0), W_ - 1);
    const int y1i = min(y0 + 1, H_ - 1);
    const int x1i = min(x0 + 1, W_ - 1);

    const float ly = ys - (float)y0;
    const float lx = xs - (float)x0;
    const float hy = 1.0f - ly;
    const float hx = 1.0f - lx;

    const f4 v00 = img[(y0  * W_ + x0 ) * 64 + c4q];
    const f4 v01 = img[(y0  * W_ + x1i) * 64 + c4q];
    const f4 v10 = img[(y1i * W_ + x0 ) * 64 + c4q];
    const f4 v11 = img[(y1i * W_ + x1i) * 64 + c4q];

    const f4 r = v00 * (hy * hx) + v01 * (hy * lx) +
                 v10 * (ly * hx) + v11 * (ly * lx);
    Xrow[bin * 64 + c4q] = r;
  }
}

// ===========================================================================
// Kernel 2: fused MLP per 16-row tile (22 blocks x 256 threads = 8 wave32s)
//   GEMM1: [16x4096]@[4096x128]  -- 8 waves, K staged 64-wide through LDS
//   GEMM2: [16x128]@[128x64]     -- waves 0..3
//   GEMM3: [16x64]@[64x10->16]   -- wave 0, zero-padded columns
//   softmax: 16 threads, one row each
// WMMA fragment mapping (ISA 7.12.2, 32-bit 16x16x4):
//   A: lane m=lane%16; vgpr0=A[m][kb], vgpr1=A[m][kb+1], kb=2*(lane>=16)
//   B: lane n=lane%16; vgpr0=B[kb][n], vgpr1=B[kb+1][n]
//   C/D: vgpr v -> row (v + 8*(lane>=16)), col n=lane%16
// ===========================================================================
__global__ __launch_bounds__(256) void mlp_head_kernel(
    const float* __restrict__ X,
    const float* __restrict__ W1, const float* __restrict__ b1,
    const float* __restrict__ W2, const float* __restrict__ b2,
    const float* __restrict__ W3, const float* __restrict__ b3,
    float* __restrict__ out) {
  // LDS (~51KB total): padded rows keep 16B alignment (stride%16B==0) while
  // shifting bank phase by 4 dwords per row.
  __shared__ __align__(16) float Xs[16][68];    // 16x64 X chunk
  __shared__ __align__(16) float Ws[64][132];   // 64x128 W1 chunk
  __shared__ __align__(16) float Hs1[16][132];  // relu(X@W1+b1)
  __shared__ __align__(16) float Hs2[16][68];   // relu(H1@W2+b2)
  __shared__ __align__(16) float Lg[16][16];    // logits (cols 10..15 junk)

  const int t    = threadIdx.x;
  const int wave = t >> 5;          // 0..7
  const int lane = t & 31;
  const int m    = lane & 15;       // A row / D col index
  const int half = lane >> 4;       // lane half
  const int kb   = half << 1;       // k-pair base: 0 or 2
  const int row0 = blockIdx.x * 16;

  // ---------------- GEMM1: acc[16x16] per wave, N-tile = wave ----------------
  const int n1 = wave * 16 + m;     // 0..127
  v8f acc = {};
  for (int kc = 0; kc < FEAT_; kc += 64) {
    // cooperative load: W1 chunk 64x128 (2048 f4, 8 per thread), coalesced
#pragma unroll
    for (int j = 0; j < 8; ++j) {
      const int idx = t + j * 256;            // 0..2047
      const int kk  = idx >> 5;               // 0..63
      const int nn  = (idx & 31) << 2;        // 0..124
      *(f4*)&Ws[kk][nn] = *(const f4*)&W1[(size_t)(kc + kk) * 128 + nn];
    }
    // X chunk 16x64 (256 f4, 1 per thread)
    {
      const int rr = t >> 4;                  // 0..15
      const int cc = (t & 15) << 2;           // 0..60
      *(f4*)&Xs[rr][cc] = *(const f4*)&X[(size_t)(row0 + rr) * FEAT_ + kc + cc];
    }
    __syncthreads();
#pragma unroll
    for (int kk = 0; kk < 64; kk += 4) {
      v2f a;  a.x  = Xs[m][kk + kb];      a.y  = Xs[m][kk + kb + 1];
      v2f bb; bb.x = Ws[kk + kb][n1];     bb.y = Ws[kk + kb + 1][n1];
      acc = __builtin_amdgcn_wmma_f32_16x16x4_f32(
          false, a, false, bb, (short)0, acc, false, false);
    }
    __syncthreads();
  }
  {
    const float bias = b1[n1];
#pragma unroll
    for (int v = 0; v < 8; ++v) {
      const float h = acc[v] + bias;
      Hs1[v + 8 * half][n1] = h > 0.0f ? h : 0.0f;
    }
  }
  __syncthreads();

  // ---------------- GEMM2: waves 0..3, N=64, K=128 (W2 from global/L2) -------
  if (wave < 4) {
    const int n2 = wave * 16 + m;   // 0..63
    v8f acc2 = {};
#pragma unroll 4
    for (int k = 0; k < 128; k += 4) {
      v2f a;  a.x  = Hs1[m][k + kb];          a.y  = Hs1[m][k + kb + 1];
      v2f bb; bb.x = W2[(k + kb) * 64 + n2];  bb.y = W2[(k + kb + 1) * 64 + n2];
      acc2 = __builtin_amdgcn_wmma_f32_16x16x4_f32(
          false, a, false, bb, (short)0, acc2, false, false);
    }
    const float bias = b2[n2];
#pragma unroll
    for (int v = 0; v < 8; ++v) {
      const float h = acc2[v] + bias;
      Hs2[v + 8 * half][n2] = h > 0.0f ? h : 0.0f;
    }
  }
  __syncthreads();

  // ---------------- GEMM3: wave 0, N padded 10->16, K=64 ---------------------
  if (wave == 0) {
    v8f acc3 = {};
#pragma unroll
    for (int k = 0; k < 64; k += 4) {
      v2f a;  a.x = Hs2[m][k + kb];  a.y = Hs2[m][k + kb + 1];
      v2f bb;
      bb.x = (m < NCLS) ? W3[(k + kb) * NCLS + m]     : 0.0f;
      bb.y = (m < NCLS) ? W3[(k + kb + 1) * NCLS + m] : 0.0f;
      acc3 = __builtin_amdgcn_wmma_f32_16x16x4_f32(
          false, a, false, bb, (short)0, acc3, false, false);
    }
    const float bias = (m < NCLS) ? b3[m] : 0.0f;
#pragma unroll
    for (int v = 0; v < 8; ++v)
      Lg[v + 8 * half][m] = acc3[v] + bias;
  }
  __syncthreads();

  // ---------------- softmax: 16 threads, one row each -------------------------
  if (t < 16) {
    float vals[NCLS];
    float mx = -INFINITY;
#pragma unroll
    for (int c = 0; c < NCLS; ++c) { vals[c] = Lg[t][c]; mx = fmaxf(mx, vals[c]); }
    float s = 0.0f;
#pragma unroll
    for (int c = 0; c < NCLS; ++c) { vals[c] = expf(vals[c] - mx); s += vals[c]; }
    const float inv = 1.0f / s;
#pragma unroll
    for (int c = 0; c < NCLS; ++c)
      out[(size_t)(row0 + t) * NCLS + c] = vals[c] * inv;
  }
}

// ===========================================================================
extern "C" void kernel_launch(void* const* d_in, const int* in_sizes, int n_in,
                              void* d_out, int out_size, void* d_ws, size_t ws_size,
                              hipStream_t stream) {
  const float* fm   = (const float*)d_in[0];   // [16,128,128,256]
  const float* prop = (const float*)d_in[1];   // [16,22,4]
  const float* W1   = (const float*)d_in[2];   // [4096,128]
  const float* b1   = (const float*)d_in[3];   // [128]
  const float* W2   = (const float*)d_in[4];   // [128,64]
  const float* b2   = (const float*)d_in[5];   // [64]
  const float* W3   = (const float*)d_in[6];   // [64,10]
  const float* b3   = (const float*)d_in[7];   // [10]
  float* out = (float*)d_out;                  // [16,22,10]
  float* X   = (float*)d_ws;                   // [352,4096] = 5.77 MB scratch

  roi_align_kernel<<<NROWS, 256, 0, stream>>>(fm, prop, X);
  mlp_head_kernel<<<NROWS / 16, 256, 0, stream>>>(X, W1, b1, W2, b2, W3, b3, out);
}